// SIM_31301721653501
// MI455X (gfx1250) — compile-verified
//
#include <hip/hip_runtime.h>
#include <hip/hip_bf16.h>
#include <math.h>

typedef float v2f __attribute__((ext_vector_type(2)));
typedef float v8f __attribute__((ext_vector_type(8)));

#define Hh 256
#define Dd 128
#define Kk 4
#define LDI 132   // padded item row stride (floats); 528B = 33*16 -> 16B aligned rows

// Async global->LDS copy of 16 bytes per lane (CDNA5 GLOBAL_LOAD_ASYNC_TO_LDS_B128,
// tracked by ASYNCcnt; no VGPR round-trip).
__device__ __forceinline__ void async_copy_b128(unsigned lds_addr, unsigned long long gaddr) {
    asm volatile("global_load_async_to_lds_b128 %0, %1, off"
                 :: "v"(lds_addr), "v"(gaddr)
                 : "memory");
}
__device__ __forceinline__ void wait_async0() {
    asm volatile("s_wait_asynccnt 0x0" ::: "memory");
}

// ---------------------------------------------------------------------------
// Kernel 1: per-sample interest extraction.
// One block = one sample. Stream all 256 gathered item vectors (128 KB) into
// LDS via async loads (320 KB/WGP on CDNA5), then scores -> softmax ->
// pooling -> argmax/select, writing x = [selected | cand] to workspace.
// ---------------------------------------------------------------------------
__global__ __launch_bounds__(256) void interest_kernel(
    const int*   __restrict__ cand_ids,   // (B)
    const int*   __restrict__ hist,       // (B,H)
    const float* __restrict__ table,      // (NUM_ITEMS, D)
    const float* __restrict__ Wp,         // (D,K) row-major
    float*       __restrict__ Xout)       // (B, 2D)
{
    extern __shared__ float smem[];
    float* items = smem;                  // 256*132
    float* wp    = items + Hh * LDI;      // 512
    float* sc    = wp + Dd * Kk;          // 1024 (scores -> softmax weights in place)
    float* cand  = sc + Hh * Kk;          // 128
    float* intv  = cand + Dd;             // 512 (K interest vectors)
    float* red   = intv + Kk * Dd;        // 16: [0..3]=max [4..7]=1/sum [8..11]=sim [12]=best

    const int b    = blockIdx.x;
    const int t    = threadIdx.x;
    const int wave = t >> 5;
    const int lane = t & 31;

    // ---- stage Wp and candidate vector ----
    for (int i = t; i < Dd * Kk; i += 256) wp[i] = Wp[i];
    {
        const int ci = cand_ids[b];
        for (int i = t; i < Dd; i += 256) cand[i] = table[(size_t)ci * Dd + i];
    }
    // ---- async gather of history rows into LDS: wave w owns rows [32w, 32w+32),
    //      each row streamed as 32 lanes x 16B (512B per instruction) ----
    const int* hrow = hist + (size_t)b * Hh;
    const unsigned items_base = (unsigned)(uintptr_t)items;   // LDS byte offset (addr[31:0])
    for (int r = 0; r < 32; ++r) {
        const int h   = wave * 32 + r;
        const int idx = hrow[h];                              // uniform per wave
        const unsigned long long gaddr =
            (unsigned long long)(uintptr_t)(table + (size_t)idx * Dd) + (unsigned)(lane * 16);
        const unsigned laddr = items_base + (unsigned)((h * LDI + lane * 4) * 4);
        async_copy_b128(laddr, gaddr);
    }
    wait_async0();
    __syncthreads();

    // ---- scores (H x K): thread t owns history row h=t ----
    {
        const int h = t;
        float a0 = 0.f, a1 = 0.f, a2 = 0.f, a3 = 0.f;
        const float* row = items + h * LDI;
        for (int d = 0; d < Dd; ++d) {
            const float x = row[d];
            const float* w4 = wp + d * Kk;                   // same addr all lanes -> broadcast
            a0 = fmaf(x, w4[0], a0);
            a1 = fmaf(x, w4[1], a1);
            a2 = fmaf(x, w4[2], a2);
            a3 = fmaf(x, w4[3], a3);
        }
        float* s4 = sc + h * Kk;
        s4[0] = a0; s4[1] = a1; s4[2] = a2; s4[3] = a3;
    }
    __syncthreads();

    // ---- softmax over H per interest head: wave k reduces column k ----
    if (wave < Kk) {
        const int k = wave;
        float m = -1e30f;
        for (int j = 0; j < 8; ++j) m = fmaxf(m, sc[(lane + 32 * j) * Kk + k]);
        for (int o = 16; o > 0; o >>= 1) m = fmaxf(m, __shfl_xor(m, o, 32));
        float s = 0.f;
        for (int j = 0; j < 8; ++j) s += __expf(sc[(lane + 32 * j) * Kk + k] - m);
        for (int o = 16; o > 0; o >>= 1) s += __shfl_xor(s, o, 32);
        if (lane == 0) { red[k] = m; red[4 + k] = 1.0f / s; }
    }
    __syncthreads();
    {   // scores -> softmax weights in place
        const int h = t;
        float* s4 = sc + h * Kk;
        s4[0] = __expf(s4[0] - red[0]) * red[4];
        s4[1] = __expf(s4[1] - red[1]) * red[5];
        s4[2] = __expf(s4[2] - red[2]) * red[6];
        s4[3] = __expf(s4[3] - red[3]) * red[7];
    }
    __syncthreads();

    // ---- weighted pooling: interest[k][d] = sum_h w[h][k]*item[h][d] ----
    {
        const int d  = t & 127;
        const int kb = t >> 7;                 // 0 or 1
        float p0 = 0.f, p1 = 0.f;
        for (int h = 0; h < Hh; ++h) {
            const float x = items[h * LDI + d];
            p0 = fmaf(sc[h * Kk + kb],     x, p0);
            p1 = fmaf(sc[h * Kk + kb + 2], x, p1);
        }
        intv[kb * Dd + d]       = p0;
        intv[(kb + 2) * Dd + d] = p1;
    }
    __syncthreads();

    // ---- similarity to candidate, argmax over K=4 ----
    if (wave < Kk) {
        const int k = wave;
        float s = 0.f;
        for (int j = 0; j < 4; ++j) {
            const int d = lane + 32 * j;
            s = fmaf(intv[k * Dd + d], cand[d], s);
        }
        for (int o = 16; o > 0; o >>= 1) s += __shfl_xor(s, o, 32);
        if (lane == 0) red[8 + k] = s;
    }
    __syncthreads();
    if (t == 0) {
        int best = 0; float bs = red[8];
        for (int k = 1; k < Kk; ++k)
            if (red[8 + k] > bs) { bs = red[8 + k]; best = k; }
        red[12] = __int_as_float(best);
    }
    __syncthreads();
    {
        const int best = __float_as_int(red[12]);
        float* xrow = Xout + (size_t)b * (2 * Dd);
        if (t < Dd)            xrow[t] = intv[best * Dd + t];
        else                   xrow[t] = cand[t - Dd];
    }
}

// ---------------------------------------------------------------------------
// Weight transpose (one-off, 1 MB total): lets the GEMM B-fragment be a single
// aligned b64 load instead of two strided b32 loads.
// W (R x C) -> WT (C x R)
// ---------------------------------------------------------------------------
__global__ __launch_bounds__(256) void transpose_kernel(
    const float* __restrict__ W, float* __restrict__ WT, int R, int C)
{
    const int i = blockIdx.x * 256 + threadIdx.x;
    if (i < R * C) {
        const int n = i / R;       // output row
        const int k = i % R;       // output col (coalesced writes)
        WT[(size_t)n * R + k] = W[(size_t)k * C + n];
    }
}

// ---------------------------------------------------------------------------
// Kernel 2: fused MLP tail with f32 WMMA (V_WMMA_F32_16X16X4_F32).
// One block = 16 batch rows; activations live in LDS; weights stay L2-hot.
// ---------------------------------------------------------------------------
#define LD_X  260   // 16x256 X tile (reused for Y2); stride%64==4 -> conflict-free A loads
#define LD_Y1 516   // 16x512 hidden-1 tile

__global__ __launch_bounds__(256) void mlp_kernel(
    const float* __restrict__ X,     // (B, 256)
    const float* __restrict__ W1T,   // (512, 256)  = W1^T
    const float* __restrict__ b1,    // (512)
    const float* __restrict__ W2T,   // (256, 512)  = W2^T
    const float* __restrict__ b2,    // (256)
    const float* __restrict__ W3,    // (256, 1)
    const float* __restrict__ b3,    // (1)
    float*       __restrict__ out)   // (B)
{
    __shared__ float Xs[16 * LD_X];     // X tile, later reused as Y2 (16x256)
    __shared__ float Y1s[16 * LD_Y1];   // hidden-1 (16x512)
    __shared__ float part[256];

    const int t    = threadIdx.x;
    const int wave = t >> 5;
    const int lid  = t & 31;
    const int m0   = blockIdx.x * 16;

    // ---- async stage of X tile (16x256 f32): 1024 x b128, 4 per thread ----
    {
        const unsigned xs_base = (unsigned)(uintptr_t)Xs;
        for (int i = t; i < 16 * 64; i += 256) {
            const int r = i >> 6, c4 = i & 63;
            const unsigned long long gaddr =
                (unsigned long long)(uintptr_t)(X + (size_t)(m0 + r) * 256 + c4 * 4);
            const unsigned laddr = xs_base + (unsigned)((r * LD_X + c4 * 4) * 4);
            async_copy_b128(laddr, gaddr);
        }
        wait_async0();
    }
    __syncthreads();

    // WMMA fragment indexing (wave32):
    //  A 16x4 f32: lanes 0-15 -> M=lane, K={k0,k0+1}; lanes 16-31 -> M=lane-16, K={k0+2,k0+3}
    //  B 4x16 f32: mirrored; C 16x16: VGPR i -> rows i / i+8 across lane halves.
    const int mrow  = lid & 15;
    const int khalf = (lid >> 4) << 1;       // 0 or 2
    const int rbase = (lid >> 4) << 3;       // 0 or 8

    // ---- layer 1: (16x256)@(256x512), ReLU -> Y1s.  8 waves x 4 N-subtiles ----
    for (int s = 0; s < 4; ++s) {
        const int n0   = (wave * 4 + s) * 16;
        const int ncol = n0 + (lid & 15);
        const float* wrow = W1T + (size_t)ncol * 256;    // column ncol of W1
        v8f acc = {0.f, 0.f, 0.f, 0.f, 0.f, 0.f, 0.f, 0.f};
        for (int k0 = 0; k0 < 256; k0 += 4) {
            const int kb = k0 + khalf;
            v2f a = *(const v2f*)(Xs + mrow * LD_X + kb);
            v2f bf = *(const v2f*)(wrow + kb);           // single aligned b64 load
            acc = __builtin_amdgcn_wmma_f32_16x16x4_f32(
                false, a, false, bf, (short)0, acc, false, false);
        }
        const float bias = b1[ncol];
        for (int i = 0; i < 8; ++i)
            Y1s[(rbase + i) * LD_Y1 + ncol] = fmaxf(acc[i] + bias, 0.0f);
    }
    __syncthreads();

    // ---- layer 2: (16x512)@(512x256), ReLU -> Y2 (reusing Xs).  2 subtiles/wave ----
    for (int s = 0; s < 2; ++s) {
        const int n0   = (wave * 2 + s) * 16;
        const int ncol = n0 + (lid & 15);
        const float* wrow = W2T + (size_t)ncol * 512;    // column ncol of W2
        v8f acc = {0.f, 0.f, 0.f, 0.f, 0.f, 0.f, 0.f, 0.f};
        for (int k0 = 0; k0 < 512; k0 += 4) {
            const int kb = k0 + khalf;
            v2f a = *(const v2f*)(Y1s + mrow * LD_Y1 + kb);
            v2f bf = *(const v2f*)(wrow + kb);
            acc = __builtin_amdgcn_wmma_f32_16x16x4_f32(
                false, a, false, bf, (short)0, acc, false, false);
        }
        const float bias = b2[ncol];
        for (int i = 0; i < 8; ++i)
            Xs[(rbase + i) * LD_X + ncol] = fmaxf(acc[i] + bias, 0.0f);
    }
    __syncthreads();

    // ---- layer 3: (16x256)@(256x1) + sigmoid ----
    {
        const int m = t >> 4, j = t & 15;
        float s = 0.f;
        for (int c = 0; c < 16; ++c) {
            const int d = j * 16 + c;
            s = fmaf(Xs[m * LD_X + d], W3[d], s);
        }
        part[t] = s;
    }
    __syncthreads();
    if (t < 16) {
        float s = 0.f;
        for (int j = 0; j < 16; ++j) s += part[t * 16 + j];
        s += b3[0];
        out[m0 + t] = 1.0f / (1.0f + __expf(-s));
    }
}

// ---------------------------------------------------------------------------
extern "C" void kernel_launch(void* const* d_in, const int* in_sizes, int n_in,
                              void* d_out, int out_size, void* d_ws, size_t ws_size,
                              hipStream_t stream) {
    // setup_inputs order:
    // 0 user_ids, 1 candidate_items, 2 history_items, 3 item_table, 4 Wp,
    // 5 W1, 6 b1, 7 W2, 8 b2, 9 W3, 10 b3
    const int*   cand  = (const int*)d_in[1];
    const int*   hist  = (const int*)d_in[2];
    const float* table = (const float*)d_in[3];
    const float* Wp    = (const float*)d_in[4];
    const float* W1    = (const float*)d_in[5];
    const float* b1    = (const float*)d_in[6];
    const float* W2    = (const float*)d_in[7];
    const float* b2    = (const float*)d_in[8];
    const float* W3    = (const float*)d_in[9];
    const float* b3    = (const float*)d_in[10];
    float* out = (float*)d_out;

    const int B = in_sizes[0];                  // 4096

    // workspace layout: X (B x 256) | W1T (512x256) | W2T (256x512)
    float* Xbuf = (float*)d_ws;
    float* W1T  = Xbuf + (size_t)B * 256;
    float* W2T  = W1T + 512 * 256;

    transpose_kernel<<<(256 * 512 + 255) / 256, 256, 0, stream>>>(W1, W1T, 256, 512);
    transpose_kernel<<<(512 * 256 + 255) / 256, 256, 0, stream>>>(W2, W2T, 512, 256);

    const size_t smem1 = (size_t)(Hh * LDI + Dd * Kk + Hh * Kk + Dd + Kk * Dd + 16) * sizeof(float);
    interest_kernel<<<B, 256, smem1, stream>>>(cand, hist, table, Wp, Xbuf);
    mlp_kernel<<<B / 16, 256, 0, stream>>>(Xbuf, W1T, b1, W2T, b2, W3, b3, out);
}